// SpinConvSq2dSparse_12369505813006
// MI455X (gfx1250) — compile-verified
//
#include <hip/hip_runtime.h>
#include <hip/hip_bf16.h>
#include <stdint.h>

typedef float v2f __attribute__((ext_vector_type(2)));
typedef float v8f __attribute__((ext_vector_type(8)));
typedef int vs4i __attribute__((__vector_size__(16)));   // matches builtin param

// D(16x16 f32) = A(16x4 f32) * B(4x16 f32) + C  -- exact fp32, wave32 WMMA.
// neg/c_mod/reuse immediates must be literals -> macro, not inline fn.
#define WMMA_ACC(A, B, C) \
  __builtin_amdgcn_wmma_f32_16x16x4_f32(false, (A), false, (B), (short)0, (C), false, false)

__device__ constexpr float C0f = 0.28209479177387814f;
__device__ constexpr float C1f = 0.4886025119029199f;
__device__ constexpr float IS3 = 0.57735026918962576f;   // 1/sqrt(3)
__device__ constexpr float IS6 = 0.40824829046386302f;   // 1/sqrt(6)
__device__ constexpr float PW0 = 0.05892556509887896f;   // sqrt(1/(9*2*16))
__device__ constexpr float PW1 = 0.08333333333333333f;   // sqrt(3/(9*3*16)) = 1/12
__device__ constexpr float S00 = PW0 * C0f;
__device__ constexpr float S0V = PW0 * IS3;
__device__ constexpr float S1A = PW1 * IS3;
__device__ constexpr float S1B = PW1 * C0f * IS3;
__device__ constexpr float S1C = PW1 * IS6;

// ---------------------------------------------------------------------------
// Prep: fold the 5 duplicate center taps into tap 0, and swizzle all five
// weight matrices into the B-fragment layout for V_WMMA_F32_16X16X4_F32:
//   Wp[m][q][lane] = float2{ W_eff[kk0+0][d], W_eff[kk0+1][d] }
//   kk = t*16 + c in [0,80), kk0 = 4*q + 2*(lane>>4), d = lane&15
// (lane l, vgpr r holds B[K = 2*(l/16)+r][N = l%16] within a K=4 chunk)
// m order matches d_in: 0=w000 1=w011 2=w101 3=w110 4=w111. Size 25.6 KB.
// ---------------------------------------------------------------------------
__global__ __launch_bounds__(256) void prep_weights(
    const float* __restrict__ w000, const float* __restrict__ w011,
    const float* __restrict__ w101, const float* __restrict__ w110,
    const float* __restrict__ w111, float* __restrict__ Wp) {
  int j = blockIdx.x * 256 + threadIdx.x;
  if (j >= 5 * 20 * 32) return;
  int m = j / 640;
  int rem = j - m * 640;
  int q = rem >> 5;
  int l = rem & 31;
  const float* w = (m == 0) ? w000 : (m == 1) ? w011 : (m == 2) ? w101
                  : (m == 3) ? w110 : w111;
  int d = l & 15;
  int kk0 = 4 * q + 2 * (l >> 4);
  float o[2];
#pragma unroll
  for (int r = 0; r < 2; ++r) {
    int kk = kk0 + r;
    int t = kk >> 4;      // effective tap 0..4
    int c = kk & 15;
    float v;
    if (t == 0) {         // taps 0..4 of the original K=9 are all the center
      v = 0.f;
#pragma unroll
      for (int k = 0; k < 5; ++k) v += w[(k * 16 + c) * 16 + d];
    } else {              // taps 5..8: x-1, x+1, y-1, y+1
      v = w[((4 + t) * 16 + c) * 16 + d];
    }
    o[r] = v;
  }
  ((v2f*)Wp)[j] = (v2f){o[0], o[1]};
}

// ---------------------------------------------------------------------------
// Main kernel. Grid (16,8,8) = (Lx/8, Ly/16, N). 256 threads = 8 waves.
// LDS tile: 10x18 pixels (8x16 core + circular halo), 68 floats/pixel
// (16 scalar | 16 V0 | 16 V1 | 16 V2 | 4 pad -> bank-conflict-free A loads).
// Wave w computes core x-row w: M=16 pixels (y), N=16 channels, K'=80.
// ---------------------------------------------------------------------------
__global__ __launch_bounds__(256) void spinconv_main(
    const float* __restrict__ feat, const float* __restrict__ spin,
    const float* __restrict__ Wp, float* __restrict__ out) {
  __shared__ float ldsFeat[180 * 68];  // 48,960 B
  __shared__ float ldsY1[128 * 3];     //  1,536 B

  const int tid = threadIdx.x;
  const int bx = blockIdx.x, by = blockIdx.y, n = blockIdx.z;

  // ---- stage feat tile, de-interleaving (c,i) channels per pixel ----------
  for (int i = tid; i < 180 * 16; i += 256) {
    int pix = i >> 4, quad = i & 15;             // quad = 4 consecutive feats
    int tx = pix / 18, ty = pix - tx * 18;
    int x = (bx * 8 + tx - 1) & 127;             // circular wrap (Lx=128)
    int y = (by * 16 + ty - 1) & 127;            // circular wrap (Ly=128)
    const float* src = feat + (((n * 128 + x) * 128 + y) * 64 + quad * 4);
    if (quad < 4) {
      // scalar plane: contiguous -> async global->LDS b128 when available
      int lo = pix * 68 + quad * 4;
#if __has_builtin(__builtin_amdgcn_global_load_async_to_lds_b128)
      __builtin_amdgcn_global_load_async_to_lds_b128(
          (__attribute__((address_space(1))) vs4i*)src,
          (__attribute__((address_space(3))) vs4i*)&ldsFeat[lo], 0, 0);
#else
      float4 v = *(const float4*)src;
      *(float4*)&ldsFeat[lo] = v;
#endif
    } else {
      // vector part: feature 16+3c+i -> plane i, channel c
      float4 v = *(const float4*)src;
      int j0 = quad * 4 - 16;
#pragma unroll
      for (int e = 0; e < 4; ++e) {
        int j = j0 + e;
        int c = j / 3, comp = j - c * 3;
        ldsFeat[pix * 68 + 16 + comp * 16 + c] = ((const float*)&v)[e];
      }
    }
  }
  // ---- stage y1 = C1 * spin[...,[1,2,0]] for the 8x16 core ----------------
  if (tid < 128) {
    int cx = tid >> 4, cy = tid & 15;
    const float* s = spin + (((n * 128 + (bx * 8 + cx)) * 128 + (by * 16 + cy)) * 3);
    ldsY1[tid * 3 + 0] = C1f * s[1];
    ldsY1[tid * 3 + 1] = C1f * s[2];
    ldsY1[tid * 3 + 2] = C1f * s[0];
  }
#if __has_builtin(__builtin_amdgcn_global_load_async_to_lds_b128)
#if __has_builtin(__builtin_amdgcn_s_wait_asynccnt)
  __builtin_amdgcn_s_wait_asynccnt(0);
#else
  asm volatile("s_wait_asynccnt 0" ::: "memory");
#endif
#endif
  __syncthreads();

  // ---- per-wave GEMM: 20 K-chunks x 11 WMMAs --------------------------------
  const int wave = tid >> 5;        // core x-row (group) 0..7
  const int lane = tid & 31;
  const int mlo = lane & 15;        // A-matrix row = pixel y in group
  const int khalf = lane >> 4;      // K sub-slot (A/B: K = 2*khalf + vgpr)

  const int dxs[5] = {0, -1, 1, 0, 0};
  const int dys[5] = {0, 0, 0, -1, 1};
  int pixBase[5];
#pragma unroll
  for (int t = 0; t < 5; ++t)
    pixBase[t] = ((wave + 1 + dxs[t]) * 18 + (mlo + 1 + dys[t])) * 68 + 2 * khalf;

  const v2f* WpV = (const v2f*)Wp;

  v8f a000 = {0, 0, 0, 0, 0, 0, 0, 0}, a011 = {0, 0, 0, 0, 0, 0, 0, 0};
  v8f t110[3] = {a000, a000, a000};
  v8f t101[3] = {a000, a000, a000};
  v8f t111[3] = {a000, a000, a000};

#pragma unroll
  for (int t = 0; t < 5; ++t) {
#pragma unroll
    for (int cb = 0; cb < 4; ++cb) {
      const int q = t * 4 + cb;
      const int ab = pixBase[t] + cb * 4;
      v2f aS = *(const v2f*)&ldsFeat[ab];        // scalar channels
      v2f a0 = *(const v2f*)&ldsFeat[ab + 16];   // vector comp 0
      v2f a1 = *(const v2f*)&ldsFeat[ab + 32];   // vector comp 1
      v2f a2 = *(const v2f*)&ldsFeat[ab + 48];   // vector comp 2
      v2f b000 = WpV[(0 * 20 + q) * 32 + lane];  // coalesced 256B, L2-hot
      v2f b011 = WpV[(1 * 20 + q) * 32 + lane];
      v2f b101 = WpV[(2 * 20 + q) * 32 + lane];
      v2f b110 = WpV[(3 * 20 + q) * 32 + lane];
      v2f b111 = WpV[(4 * 20 + q) * 32 + lane];
      a000 = WMMA_ACC(aS, b000, a000);
      a011 = WMMA_ACC(aS, b011, a011);
      t110[0] = WMMA_ACC(a0, b110, t110[0]);
      t110[1] = WMMA_ACC(a1, b110, t110[1]);
      t110[2] = WMMA_ACC(a2, b110, t110[2]);
      t101[0] = WMMA_ACC(a0, b101, t101[0]);
      t101[1] = WMMA_ACC(a1, b101, t101[1]);
      t101[2] = WMMA_ACC(a2, b101, t101[2]);
      t111[0] = WMMA_ACC(a0, b111, t111[0]);
      t111[1] = WMMA_ACC(a1, b111, t111[1]);
      t111[2] = WMMA_ACC(a2, b111, t111[2]);
    }
  }

  // ---- epilogue: combine with y1, cross product, write out ----------------
  // D layout: vgpr r, lanes0-15 -> pixel r, lanes16-31 -> pixel r+8; N=lane&15
  const int d = mlo;
  const int gx = bx * 8 + wave;
#pragma unroll
  for (int r = 0; r < 8; ++r) {
    int mrow = r + 8 * khalf;
    const float* y1p = &ldsY1[(wave * 16 + mrow) * 3];  // broadcast ds reads
    float y0 = y1p[0], y1v = y1p[1], y2 = y1p[2];
    float s110 = t110[0][r] * y0 + t110[1][r] * y1v + t110[2][r] * y2;
    float o0 = S00 * a000[r] + S0V * s110;
    float bsc = S1A * a011[r];
    float u0 = t111[0][r], u1 = t111[1][r], u2 = t111[2][r];
    float cx0 = u1 * y2 - u2 * y1v;     // cross(t111, y1)
    float cx1 = u2 * y0 - u0 * y2;
    float cx2 = u0 * y1v - u1 * y0;
    float o10 = bsc * y0 + S1B * t101[0][r] + S1C * cx0;
    float o11 = bsc * y1v + S1B * t101[1][r] + S1C * cx1;
    float o12 = bsc * y2 + S1B * t101[2][r] + S1C * cx2;
    float* op = out + (((n * 128 + gx) * 128 + (by * 16 + mrow)) * 64);
    op[d] = o0;                         // out0
    op[16 + 3 * d + 0] = o10;           // out1 reshaped (d,i) -> 16+3d+i
    op[16 + 3 * d + 1] = o11;
    op[16 + 3 * d + 2] = o12;
  }
}

extern "C" void kernel_launch(void* const* d_in, const int* in_sizes, int n_in,
                              void* d_out, int out_size, void* d_ws, size_t ws_size,
                              hipStream_t stream) {
  const float* feat = (const float*)d_in[0];
  const float* spin = (const float*)d_in[1];
  const float* w000 = (const float*)d_in[2];
  const float* w011 = (const float*)d_in[3];
  const float* w101 = (const float*)d_in[4];
  const float* w110 = (const float*)d_in[5];
  const float* w111 = (const float*)d_in[6];
  float* Wp = (float*)d_ws;                 // 25,600 B swizzled weights
  float* outp = (float*)d_out;

  hipLaunchKernelGGL(prep_weights, dim3((5 * 20 * 32 + 255) / 256), dim3(256),
                     0, stream, w000, w011, w101, w110, w111, Wp);
  hipLaunchKernelGGL(spinconv_main, dim3(16, 8, 8), dim3(256), 0, stream,
                     feat, spin, Wp, outp);
}